// ContextNet_45535243272621
// MI455X (gfx1250) — compile-verified
//
#include <hip/hip_runtime.h>

typedef __attribute__((ext_vector_type(2))) float v2f;
typedef __attribute__((ext_vector_type(4))) float v4f;
typedef __attribute__((ext_vector_type(8))) float v8f;

#define BB 8
#define AA 64
#define NN 4096
#define DD 128
#define R2 36.0f        // (6.0 m)^2
#define NEGF -1.0e30f

__device__ __forceinline__ v4f vmax4(v4f a, v4f b) {
  v4f r;
  r.x = fmaxf(a.x, b.x);
  r.y = fmaxf(a.y, b.y);
  r.z = fmaxf(a.z, b.z);
  r.w = fmaxf(a.w, b.w);
  return r;
}

// One block = one (batch, 16-actor tile). 8 waves; wave w owns nodes
// [w*512, w*512+512). dist^2 tiles come from V_WMMA_F32_16X16X4_F32 with
//   A (16x4) = [-2ax, -2ay, |a|^2, 1]   (per actor row)
//   B (4x16) = [nx, ny, 1, |n|^2]^T     (per node column)
// so D[m][n] = |a_m - c_n|^2 in a single matrix op per tile. Sparse hits are
// consumed with scalar-uniform control flow (ballot -> ctz loop -> readlane).
__global__ __launch_bounds__(256) void ctxnet_wmma_kernel(
    const float* __restrict__ nodes,
    const float* __restrict__ actor_ctrs,
    const float* __restrict__ node_ctrs,
    float* __restrict__ out) {
  __shared__ v4f red[4][16][32];  // 32 KB cross-wave max buffer

  const int tid  = threadIdx.x;
  const int lane = tid & 31;
  const int w    = tid >> 5;               // wave id, 0..7
  const int b    = blockIdx.x >> 2;        // batch
  const int am0  = (blockIdx.x & 3) << 4;  // actor tile base

  // ---- WMMA A operand (uniform across the node loop) ----
  // A-matrix 16x4 f32 layout: lanes 0-15 hold (K=0,K=1), lanes 16-31 (K=2,K=3)
  const v2f* actor2 = (const v2f*)actor_ctrs;
  v2f ac = actor2[(size_t)b * AA + am0 + (lane & 15)];
  v2f Aop;
  Aop.x = (lane < 16) ? (-2.0f * ac.x) : (ac.x * ac.x + ac.y * ac.y);
  Aop.y = (lane < 16) ? (-2.0f * ac.y) : 1.0f;

  // Register-resident masked-max accumulators: lane holds features
  // f = 4*lane .. 4*lane+3 for each of the 16 actors in this tile.
  v4f acc[16];
#pragma unroll
  for (int m = 0; m < 16; ++m) acc[m] = (v4f){NEGF, NEGF, NEGF, NEGF};

  const v2f*   node2 = (const v2f*)node_ctrs;
  const float* nbasep = nodes + (size_t)b * NN * DD;
  const int    nbase  = w * (NN / 8);  // 512 nodes per wave

  for (int t = 0; t < (NN / 8) / 16; ++t) {
    const int n0 = nbase + t * 16;

    // B-matrix 4x16 f32 layout: lanes 0-15 hold (K=0,K=1) of column n=lane,
    // lanes 16-31 hold (K=2,K=3) of column n=lane-16.
    v2f nc = node2[(size_t)b * NN + n0 + (lane & 15)];
    v2f Bop;
    Bop.x = (lane < 16) ? nc.x : 1.0f;
    Bop.y = (lane < 16) ? nc.y : (nc.x * nc.x + nc.y * nc.y);

    v8f c8 = {};
    // D[m][n] = squared actor->node distance, 16x16 tile
    v8f d = __builtin_amdgcn_wmma_f32_16x16x4_f32(
        false, Aop, false, Bop, (short)0, c8, false, false);

    // Per-lane 8-bit hit mask: bit i set iff D vgpr i is within radius.
    unsigned mv = 0u;
#pragma unroll
    for (int i = 0; i < 8; ++i) mv |= (d[i] <= R2) ? (1u << i) : 0u;

    // C/D layout: lane n (n<16) carries m=0..7 of node n; lane n+16 carries
    // m=8..15. Ballot the lane-has-hit predicate and fold the two halves into
    // a uniform 16-bit per-node mask so we only visit nodes with >=1 hit.
    unsigned any = __builtin_amdgcn_ballot_w32(mv != 0u);
    unsigned hit = (any & 0xFFFFu) | (any >> 16);

    while (hit) {
      const int n = __builtin_ctz(hit);   // s_ff1 on a uniform mask
      hit &= hit - 1u;

      int mlo = __builtin_amdgcn_readlane((int)mv, n);
      int mhi = __builtin_amdgcn_readlane((int)mv, n + 16);
      int m16 = mlo | (mhi << 8);  // uniform: which of 16 actors hit node n

      const v4f* row = (const v4f*)(nbasep + (size_t)(n0 + n) * DD);
      v4f f = row[lane];           // coalesced 512B feature row
#pragma unroll
      for (int m = 0; m < 16; ++m)
        if (m16 & (1 << m)) acc[m] = vmax4(acc[m], f);  // scalar-guarded
    }
  }

  // ---- cross-wave max reduction (2 rounds to keep LDS at 32 KB) ----
  if (w < 4) {
#pragma unroll
    for (int m = 0; m < 16; ++m) red[w][m][lane] = acc[m];
  }
  __syncthreads();
  if (w >= 4) {
#pragma unroll
    for (int m = 0; m < 16; ++m)
      red[w - 4][m][lane] = vmax4(red[w - 4][m][lane], acc[m]);
  }
  __syncthreads();

  // ---- finalize: max over 4 partials, NEG->0 fixup, coalesced store ----
  v4f* out4 = (v4f*)out;
  for (int idx = tid; idx < 16 * 32; idx += 256) {
    int m = idx >> 5, l = idx & 31;
    v4f v = vmax4(vmax4(red[0][m][l], red[1][m][l]),
                  vmax4(red[2][m][l], red[3][m][l]));
    v.x = (v.x < NEGF * 0.5f) ? 0.0f : v.x;
    v.y = (v.y < NEGF * 0.5f) ? 0.0f : v.y;
    v.z = (v.z < NEGF * 0.5f) ? 0.0f : v.z;
    v.w = (v.w < NEGF * 0.5f) ? 0.0f : v.w;
    out4[(size_t)(b * AA + am0 + m) * (DD / 4) + l] = v;
  }
}

extern "C" void kernel_launch(void* const* d_in, const int* in_sizes, int n_in,
                              void* d_out, int out_size, void* d_ws, size_t ws_size,
                              hipStream_t stream) {
  const float* nodes      = (const float*)d_in[0];  // [B,N,D] f32
  const float* actor_ctrs = (const float*)d_in[1];  // [B,A,2] f32
  const float* node_ctrs  = (const float*)d_in[2];  // [B,N,2] f32
  float*       out        = (float*)d_out;          // [B*A, D] f32

  (void)in_sizes; (void)n_in; (void)out_size; (void)d_ws; (void)ws_size;

  dim3 grid(BB * (AA / 16));  // 32 blocks: (batch, actor-tile)
  dim3 block(256);            // 8 waves of 32
  ctxnet_wmma_kernel<<<grid, block, 0, stream>>>(nodes, actor_ctrs, node_ctrs, out);
}